// GCN_730144440782
// MI455X (gfx1250) — compile-verified
//
#include <hip/hip_runtime.h>

typedef float v2f __attribute__((ext_vector_type(2)));
typedef float v8f __attribute__((ext_vector_type(8)));

#define DH 128   // D_IN == D_H == 128
#define DO 40    // D_OUT

// ---------------------------------------------------------------------------
// GEMM1: m1[N,128] = x[N,128] @ W1[128,128]
// grid.x = N/16 row tiles, block = 256 (8 waves), wave w -> column tile w*16.
// K-loop: 32 x v_wmma_f32_16x16x4_f32 (exact f32, matches reference dtype).
// ---------------------------------------------------------------------------
__global__ __launch_bounds__(256)
void gcn_gemm1(const float* __restrict__ x, const float* __restrict__ W,
               float* __restrict__ out, int N) {
  const int lane = threadIdx.x & 31;
  const int wave = threadIdx.x >> 5;
  const int r0 = blockIdx.x * 16;
  const int c0 = wave * 16;
  const int m  = lane & 15;         // A row / B col / D col within tile
  const int kk = (lane >> 4) << 1;  // 0 (lanes 0-15) or 2 (lanes 16-31)

  const int arow = r0 + m;
  const float* __restrict__ xrow = x + (size_t)(arow < N ? arow : N - 1) * DH;

  v8f acc = {};
  #pragma unroll 4
  for (int k = 0; k < DH; k += 4) {
    v2f a, b;
    a.x = xrow[k + kk];
    a.y = xrow[k + kk + 1];
    b.x = W[(size_t)(k + kk) * DH + c0 + m];
    b.y = W[(size_t)(k + kk + 1) * DH + c0 + m];
    acc = __builtin_amdgcn_wmma_f32_16x16x4_f32(
        false, a, false, b, (short)0, acc, false, false);
  }

  const int rbase = r0 + ((lane >> 4) << 3);  // lanes 16-31 hold M=8..15
  const int col = c0 + m;
  float* __restrict__ p = out + (size_t)rbase * DH + col;
  if (r0 + 16 <= N) {
    // full tile: unguarded, clause-able stores (common case; N%16==0 here)
    #pragma unroll
    for (int j = 0; j < 8; ++j) p[(size_t)j * DH] = acc[j];
  } else {
    #pragma unroll
    for (int j = 0; j < 8; ++j)
      if (rbase + j < N) p[(size_t)j * DH] = acc[j];
  }
}

// ---------------------------------------------------------------------------
// GEMM2: m2[N,40] = h[N,128] @ W2[128,40]
// grid = (N/16, 3); col tile 2 is zero-padded 40->48. block = 32 (one wave).
// ---------------------------------------------------------------------------
__global__ __launch_bounds__(32)
void gcn_gemm2(const float* __restrict__ h, const float* __restrict__ W2,
               float* __restrict__ out, int N) {
  const int lane = threadIdx.x & 31;
  const int r0 = blockIdx.x * 16;
  const int c0 = blockIdx.y * 16;
  const int m  = lane & 15;
  const int kk = (lane >> 4) << 1;

  const int arow = r0 + m;
  const float* __restrict__ hrow = h + (size_t)(arow < N ? arow : N - 1) * DH;
  const int  col = c0 + m;
  const bool cv  = (col < DO);

  v8f acc = {};
  #pragma unroll 4
  for (int k = 0; k < DH; k += 4) {
    v2f a, b;
    a.x = hrow[k + kk];
    a.y = hrow[k + kk + 1];
    b.x = cv ? W2[(size_t)(k + kk) * DO + col] : 0.0f;
    b.y = cv ? W2[(size_t)(k + kk + 1) * DO + col] : 0.0f;
    acc = __builtin_amdgcn_wmma_f32_16x16x4_f32(
        false, a, false, b, (short)0, acc, false, false);
  }

  if (!cv) return;
  const int rbase = r0 + ((lane >> 4) << 3);
  float* __restrict__ p = out + (size_t)rbase * DO + col;
  if (r0 + 16 <= N) {
    #pragma unroll
    for (int j = 0; j < 8; ++j) p[(size_t)j * DO] = acc[j];
  } else {
    #pragma unroll
    for (int j = 0; j < 8; ++j)
      if (rbase + j < N) p[(size_t)j * DO] = acc[j];
  }
}

// ---------------------------------------------------------------------------
// Edge scatter: agg[dst,c] += m[src,c] * w[e]. One thread per (edge, feature).
// Consecutive threads cover consecutive features of one edge -> coalesced
// gathers; atomic targets are L2-resident (25.6 MB << 192 MB L2).
// ---------------------------------------------------------------------------
template <int D>
__global__ __launch_bounds__(256)
void gcn_scatter(const float* __restrict__ msrc, const int* __restrict__ src,
                 const int* __restrict__ dst, const float* __restrict__ w,
                 float* __restrict__ agg, long long total) {
  long long t = (long long)blockIdx.x * blockDim.x + threadIdx.x;
  if (t >= total) return;
  const int e = (int)(t / D);
  const int c = (int)(t % D);
  const int s = src[e];
  const int d = dst[e];
  const float v = msrc[(size_t)s * D + c] * w[e];
  atomicAdd(&agg[(size_t)d * D + c], v);
}

__global__ __launch_bounds__(256)
void gcn_zero(float* __restrict__ p, long long n) {
  long long t = (long long)blockIdx.x * blockDim.x + threadIdx.x;
  if (t < n) p[t] = 0.0f;
}

// h = relu(agg + b1), in place; D fixed at 128 so %/ are shifts.
__global__ __launch_bounds__(256)
void gcn_bias_relu(float* __restrict__ agg, const float* __restrict__ b,
                   long long n) {
  long long t = (long long)blockIdx.x * blockDim.x + threadIdx.x;
  if (t >= n) return;
  const int c = (int)(t & (DH - 1));
  const float v = agg[t] + b[c];
  agg[t] = v > 0.0f ? v : 0.0f;
}

// out[i,c] = b2[c]  (seed for the second scatter-add)
__global__ __launch_bounds__(256)
void gcn_init_bias(float* __restrict__ out, const float* __restrict__ b,
                   long long n) {
  long long t = (long long)blockIdx.x * blockDim.x + threadIdx.x;
  if (t >= n) return;
  out[t] = b[(int)(t % DO)];
}

// ---------------------------------------------------------------------------
extern "C" void kernel_launch(void* const* d_in, const int* in_sizes, int n_in,
                              void* d_out, int out_size, void* d_ws,
                              size_t ws_size, hipStream_t stream) {
  const float* x  = (const float*)d_in[0];
  const int*   ei = (const int*)d_in[1];
  const float* ew = (const float*)d_in[2];
  const float* W1 = (const float*)d_in[3];
  const float* b1 = (const float*)d_in[4];
  const float* W2 = (const float*)d_in[5];
  const float* b2 = (const float*)d_in[6];

  const int N = in_sizes[0] / DH;   // 50000
  const int E = in_sizes[1] / 2;    // 800000
  const int* src = ei;              // edge_index[0]
  const int* dst = ei + E;          // edge_index[1]

  float* m1   = (float*)d_ws;                   // [N,128]
  float* agg1 = m1 + (size_t)N * DH;            // [N,128] -> becomes h
  float* m2   = agg1 + (size_t)N * DH;          // [N,40]
  float* out  = (float*)d_out;                  // [N,40]

  const int nt = 256;
  const int rowTiles = (N + 15) / 16;

  // 1) agg1 = 0
  {
    long long n = (long long)N * DH;
    gcn_zero<<<(int)((n + nt - 1) / nt), nt, 0, stream>>>(agg1, n);
  }
  // 2) m1 = x @ W1  (WMMA f32)
  gcn_gemm1<<<rowTiles, 256, 0, stream>>>(x, W1, m1, N);
  // 3) agg1 += scatter(m1)
  {
    long long total = (long long)E * DH;
    gcn_scatter<DH><<<(int)((total + nt - 1) / nt), nt, 0, stream>>>(
        m1, src, dst, ew, agg1, total);
  }
  // 4) h = relu(agg1 + b1)
  {
    long long n = (long long)N * DH;
    gcn_bias_relu<<<(int)((n + nt - 1) / nt), nt, 0, stream>>>(agg1, b1, n);
  }
  // 5) m2 = h @ W2  (WMMA f32, padded col tiles)
  {
    dim3 g(rowTiles, 3);
    gcn_gemm2<<<g, 32, 0, stream>>>(agg1, W2, m2, N);
  }
  // 6) out = b2 (broadcast)
  {
    long long n = (long long)N * DO;
    gcn_init_bias<<<(int)((n + nt - 1) / nt), nt, 0, stream>>>(out, b2, n);
  }
  // 7) out += scatter(m2)
  {
    long long total = (long long)E * DO;
    gcn_scatter<DO><<<(int)((total + nt - 1) / nt), nt, 0, stream>>>(
        m2, src, dst, ew, out, total);
  }
}